// SpatialAbstraction_Attention_4398046511723
// MI455X (gfx1250) — compile-verified
//
#include <hip/hip_runtime.h>
#include <hip/hip_bf16.h>
#include <math.h>

// Problem constants (match reference)
#define BB 4
#define NN 8192
#define SS 2048
#define KK 32
#define DD 64
#define PP (BB * SS * KK)   // 262144 positions
#define R2 0.04f            // RADIUS^2
#define EPSV 1e-5f

typedef __attribute__((ext_vector_type(2))) float v2f;
typedef __attribute__((ext_vector_type(8))) float v8f;

// ---------------------------------------------------------------------------
// 1) Farthest point sampling: one 1024-thread workgroup per batch.
//    Points held in registers (8 per thread), LDS tree argmax per step.
// ---------------------------------------------------------------------------
#define FPS_T 1024
#define PPT (NN / FPS_T)   // 8 points per thread

__global__ void fps_kernel(const float* __restrict__ xyz,
                           int* __restrict__ fidx,
                           float* __restrict__ newxyz) {
  int b = blockIdx.x;
  int t = threadIdx.x;
  const float* xb = xyz + (size_t)b * 3 * NN;
  float px[PPT], py[PPT], pz[PPT], mind[PPT];
#pragma unroll
  for (int i = 0; i < PPT; i++) {
    int n = t + i * FPS_T;
    px[i] = xb[n]; py[i] = xb[NN + n]; pz[i] = xb[2 * NN + n];
    mind[i] = 1e10f;
  }
  __shared__ float rv[FPS_T];
  __shared__ int   ri[FPS_T];
  __shared__ float scx, scy, scz;
  __shared__ int   sfar;
  if (t == 0) sfar = 0;
  __syncthreads();

  for (int s = 0; s < SS; s++) {
    int far = sfar;
    if (t == 0) {
      float cx = xb[far], cy = xb[NN + far], cz = xb[2 * NN + far];
      scx = cx; scy = cy; scz = cz;
      fidx[b * SS + s] = far;
      newxyz[(size_t)b * 3 * SS + s]           = cx;
      newxyz[(size_t)b * 3 * SS + SS + s]      = cy;
      newxyz[(size_t)b * 3 * SS + 2 * SS + s]  = cz;
    }
    __syncthreads();
    float ccx = scx, ccy = scy, ccz = scz;
    float bv = -1.0f; int bi = 0;
#pragma unroll
    for (int i = 0; i < PPT; i++) {
      float dx = px[i] - ccx, dy = py[i] - ccy, dz = pz[i] - ccz;
      float d = dx * dx + dy * dy + dz * dz;
      mind[i] = fminf(mind[i], d);
      int n = t + i * FPS_T;
      if (mind[i] > bv || (mind[i] == bv && n < bi)) { bv = mind[i]; bi = n; }
    }
    rv[t] = bv; ri[t] = bi;
    __syncthreads();
    for (int off = FPS_T / 2; off > 0; off >>= 1) {
      if (t < off) {
        float ov = rv[t + off]; int oi = ri[t + off];
        if (ov > rv[t] || (ov == rv[t] && oi < ri[t])) { rv[t] = ov; ri[t] = oi; }
      }
      __syncthreads();
    }
    if (t == 0) sfar = ri[0];
    __syncthreads();
  }
}

// ---------------------------------------------------------------------------
// 2) Ball query: one wave32 per center. Ballot + popcount prefix collects the
//    first K qualifying indices in scan order (== reference's sort trick).
// ---------------------------------------------------------------------------
__global__ void ball_query_kernel(const float* __restrict__ xyz,
                                  const float* __restrict__ centers,
                                  int* __restrict__ gidx) {
  int gtid = blockIdx.x * blockDim.x + threadIdx.x;
  int wid = gtid >> 5, lane = gtid & 31;
  if (wid >= BB * SS) return;
  int b = wid / SS, s = wid % SS;
  const float* xb = xyz + (size_t)b * 3 * NN;
  float ccx = centers[(size_t)b * 3 * SS + s];
  float ccy = centers[(size_t)b * 3 * SS + SS + s];
  float ccz = centers[(size_t)b * 3 * SS + 2 * SS + s];
  int* row = gidx + (size_t)wid * KK;
  int cnt = 0, first = 0;
  for (int n0 = 0; n0 < NN; n0 += 32) {
    int n = n0 + lane;
    float dx = xb[n] - ccx, dy = xb[NN + n] - ccy, dz = xb[2 * NN + n] - ccz;
    float d2 = dx * dx + dy * dy + dz * dz;
    unsigned mask = (unsigned)__ballot(d2 < R2);
    if (mask) {
      if (cnt == 0) first = n0 + __builtin_ctz(mask);
      if (d2 < R2) {
        int pos = cnt + __popc(mask & ((1u << lane) - 1u));
        if (pos < KK) row[pos] = n;
      }
      cnt += __popc(mask);
      if (cnt >= KK) break;
    }
  }
  if (lane >= cnt && lane < KK) row[lane] = (cnt > 0) ? first : 0;
}

// ---------------------------------------------------------------------------
// 3) Grouping kernels: build X[c][p] feature matrices (channel-major so GEMM
//    B-operand loads are coalesced). Pad rows zeroed.
// ---------------------------------------------------------------------------
__global__ void group1_kernel(const float* __restrict__ xyz,
                              const float* __restrict__ points,
                              const int* __restrict__ fidx,
                              const int* __restrict__ gidx,
                              const float* __restrict__ newxyz,
                              float* __restrict__ X, int CINP) {
  int p = blockIdx.x * blockDim.x + threadIdx.x;
  if (p >= PP) return;
  int bs = p / KK; int s = bs % SS; int b = bs / SS;
  int gi = gidx[p];
  int fi = fidx[bs];
  const float* xb = xyz + (size_t)b * 3 * NN;
  float cx = newxyz[(size_t)b * 3 * SS + s];
  float cy = newxyz[(size_t)b * 3 * SS + SS + s];
  float cz = newxyz[(size_t)b * 3 * SS + 2 * SS + s];
  X[(size_t)0 * PP + p] = xb[gi] - cx;
  X[(size_t)1 * PP + p] = xb[NN + gi] - cy;
  X[(size_t)2 * PP + p] = xb[2 * NN + gi] - cz;
  const float* pb = points + (size_t)b * DD * NN;
  for (int d = 0; d < DD; d++)
    X[(size_t)(3 + d) * PP + p] = pb[(size_t)d * NN + gi];
  for (int d = 0; d < DD; d++)
    X[(size_t)(3 + DD + d) * PP + p] = pb[(size_t)d * NN + fi];
  for (int c = 3 + 2 * DD; c < CINP; c++)
    X[(size_t)c * PP + p] = 0.0f;
}

__global__ void group2_kernel(const float* __restrict__ xyz,
                              const float* __restrict__ points,
                              const int* __restrict__ gidx1,
                              const float* __restrict__ centers1,
                              float* __restrict__ X, int CINP) {
  int p = blockIdx.x * blockDim.x + threadIdx.x;
  if (p >= PP) return;
  int bs = p / KK; int s = bs % SS; int b = bs / SS;
  int gi = gidx1[p];
  const float* xb = xyz + (size_t)b * 3 * NN;
  float cx = centers1[(size_t)b * 3 * SS + s];
  float cy = centers1[(size_t)b * 3 * SS + SS + s];
  float cz = centers1[(size_t)b * 3 * SS + 2 * SS + s];
  X[(size_t)0 * PP + p] = xb[gi] - cx;
  X[(size_t)1 * PP + p] = xb[NN + gi] - cy;
  X[(size_t)2 * PP + p] = xb[2 * NN + gi] - cz;
  const float* pb = points + (size_t)b * DD * NN;
  for (int d = 0; d < DD; d++)
    X[(size_t)(3 + d) * PP + p] = pb[(size_t)d * NN + gi];
  for (int c = 3 + DD; c < CINP; c++)
    X[(size_t)c * PP + p] = 0.0f;
}

// ---------------------------------------------------------------------------
// 4a) Weight pre-pad: Wp[CO][CINP] = W zero-extended. Removes all divergent
//     guards from the GEMM inner loop and makes A fragments one b64 load.
// ---------------------------------------------------------------------------
__global__ void pad_weights_kernel(const float* __restrict__ W,
                                   float* __restrict__ Wp,
                                   int CO, int CIN, int CINP) {
  int i = blockIdx.x * blockDim.x + threadIdx.x;
  if (i >= CO * CINP) return;
  int o = i / CINP, k = i % CINP;
  Wp[i] = (k < CIN) ? W[(size_t)o * CIN + k] : 0.0f;
}

// ---------------------------------------------------------------------------
// 4b) fp32 GEMM on the WMMA pipe: Y[o,p] = sum_c Wp[o,c]*X[c,p].
//     One wave per 16x64 output tile (4 accumulators share one A fragment);
//     V_WMMA_F32_16X16X4_F32, fully unrolled k-loop (CINP compile-time).
//     A layout: rows M=lane&15; lanes 0-15 hold K=kc..kc+1, lanes 16-31 hold
//     K=kc+2..kc+3 (consecutive in the padded row -> single b64 load).
//     C/D layout: row = v + 8*(lane>=16), col = lane&15.
// ---------------------------------------------------------------------------
template <int CINP>
__global__ void gemm_wmma_kernel(const float* __restrict__ Wp,
                                 const float* __restrict__ X,
                                 float* __restrict__ Y, int CO) {
  int gtid = blockIdx.x * blockDim.x + threadIdx.x;
  int wid = gtid >> 5, lane = gtid & 31;
  int tilesM = CO >> 4;
  int tilesN = PP >> 6;                 // 64 columns per wave
  if (wid >= tilesM * tilesN) return;
  int m0 = (wid % tilesM) << 4;
  int n0 = (wid / tilesM) << 6;
  int hi = lane >> 4, lm = lane & 15;

  const float* wrow = Wp + (size_t)(m0 + lm) * CINP + 2 * hi;
  const float* xcol = X + (size_t)(2 * hi) * PP + n0 + lm;

  v8f acc0 = {}, acc1 = {}, acc2 = {}, acc3 = {};
#pragma unroll
  for (int kc = 0; kc < CINP; kc += 4) {
    v2f a = *(const v2f*)(wrow + kc);               // W[m][kc+2hi .. +1]
    const float* xk0 = xcol + (size_t)kc * PP;      // X[kc+2hi][n]
    const float* xk1 = xk0 + PP;                    // X[kc+2hi+1][n]
    if (kc + 4 < CINP)
      __builtin_prefetch(xk0 + (size_t)4 * PP, 0, 1);
    v2f b0, b1, b2, b3;
    b0.x = xk0[0];  b0.y = xk1[0];
    b1.x = xk0[16]; b1.y = xk1[16];
    b2.x = xk0[32]; b2.y = xk1[32];
    b3.x = xk0[48]; b3.y = xk1[48];
    acc0 = __builtin_amdgcn_wmma_f32_16x16x4_f32(false, a, false, b0, (short)0, acc0, false, false);
    acc1 = __builtin_amdgcn_wmma_f32_16x16x4_f32(false, a, false, b1, (short)0, acc1, false, false);
    acc2 = __builtin_amdgcn_wmma_f32_16x16x4_f32(false, a, false, b2, (short)0, acc2, false, false);
    acc3 = __builtin_amdgcn_wmma_f32_16x16x4_f32(false, a, false, b3, (short)0, acc3, false, false);
  }
#pragma unroll
  for (int v = 0; v < 8; v++) {
    size_t r = (size_t)(m0 + v + hi * 8) * PP + n0 + lm;
    Y[r]      = acc0[v];
    Y[r + 16] = acc1[v];
    Y[r + 32] = acc2[v];
    Y[r + 48] = acc3[v];
  }
}

// ---------------------------------------------------------------------------
// 5) Per-channel batch statistics (training-mode BN) + normalize/ReLU.
// ---------------------------------------------------------------------------
__global__ void chan_stats_kernel(const float* __restrict__ Y,
                                  float* __restrict__ mean,
                                  float* __restrict__ rstd) {
  int o = blockIdx.x, t = threadIdx.x;
  const float* row = Y + (size_t)o * PP;
  float s = 0.0f, s2 = 0.0f;
  for (int p = t; p < PP; p += 256) { float v = row[p]; s += v; s2 += v * v; }
  __shared__ float sh[256], sh2[256];
  sh[t] = s; sh2[t] = s2;
  __syncthreads();
  for (int off = 128; off > 0; off >>= 1) {
    if (t < off) { sh[t] += sh[t + off]; sh2[t] += sh2[t + off]; }
    __syncthreads();
  }
  if (t == 0) {
    float m = sh[0] / (float)PP;
    float var = sh2[0] / (float)PP - m * m;
    mean[o] = m;
    rstd[o] = rsqrtf(var + EPSV);
  }
}

__global__ void bn_relu_kernel(const float* __restrict__ Y,
                               const float* __restrict__ mean,
                               const float* __restrict__ rstd,
                               const float* __restrict__ gg,
                               const float* __restrict__ bb,
                               float* __restrict__ Xo, int CO) {
  size_t i = (size_t)blockIdx.x * blockDim.x + threadIdx.x;
  size_t tot = (size_t)CO * PP;
  if (i >= tot) return;
  int c = (int)(i / PP);
  float v = (Y[i] - mean[c]) * rstd[c] * gg[c] + bb[c];
  Xo[i] = v > 0.0f ? v : 0.0f;
}

// ---------------------------------------------------------------------------
// 6) Attention pooling: np1 = max_k x; scores = sum_c x*np1; softmax over K
//    (one wave, shuffle reductions); weighted centroid of gathered xyz.
// ---------------------------------------------------------------------------
__global__ void attention_kernel(const float* __restrict__ X,
                                 const int* __restrict__ gidx,
                                 const float* __restrict__ xyz,
                                 float* __restrict__ newxyz1) {
  int bs = blockIdx.x;
  int b = bs / SS, s = bs % SS;
  int c = threadIdx.x;           // 128 threads = 128 channels
  size_t p0 = (size_t)bs * KK;
  __shared__ float np1[128];
  const float* row = X + (size_t)c * PP + p0;
  float mx = -3.4e38f;
  for (int k = 0; k < KK; k++) mx = fmaxf(mx, row[k]);
  np1[c] = mx;
  __syncthreads();
  if (c < KK) {
    float sco = 0.0f;
    for (int cc = 0; cc < 128; cc++) sco += X[(size_t)cc * PP + p0 + c] * np1[cc];
    float m = sco;
    for (int off = 16; off > 0; off >>= 1) m = fmaxf(m, __shfl_xor(m, off, 32));
    float e = expf(sco - m);
    float sum = e;
    for (int off = 16; off > 0; off >>= 1) sum += __shfl_xor(sum, off, 32);
    float prob = e / sum;
    int gi = gidx[p0 + c];
    const float* xb = xyz + (size_t)b * 3 * NN;
    float wx = xb[gi] * prob, wy = xb[NN + gi] * prob, wz = xb[2 * NN + gi] * prob;
    for (int off = 16; off > 0; off >>= 1) {
      wx += __shfl_xor(wx, off, 32);
      wy += __shfl_xor(wy, off, 32);
      wz += __shfl_xor(wz, off, 32);
    }
    if (c == 0) {
      newxyz1[(size_t)b * 3 * SS + s]          = wx;
      newxyz1[(size_t)b * 3 * SS + SS + s]     = wy;
      newxyz1[(size_t)b * 3 * SS + 2 * SS + s] = wz;
    }
  }
}

// ---------------------------------------------------------------------------
// 7) Final max over K -> (B,128,S)
// ---------------------------------------------------------------------------
__global__ void maxk_kernel(const float* __restrict__ X, float* __restrict__ out) {
  int i = blockIdx.x * blockDim.x + threadIdx.x;   // over B*128*S
  if (i >= BB * 128 * SS) return;
  int s = i % SS; int bc = i / SS; int c = bc % 128; int b = bc / 128;
  size_t p0 = (size_t)(b * SS + s) * KK;
  const float* row = X + (size_t)c * PP + p0;
  float mx = row[0];
  for (int k = 1; k < KK; k++) mx = fmaxf(mx, row[k]);
  out[((size_t)b * 128 + c) * SS + s] = mx;
}

// ---------------------------------------------------------------------------
// Launch
// ---------------------------------------------------------------------------
static inline size_t align256(size_t x) { return (x + 255) & ~(size_t)255; }

static void conv_bn_relu(const float* W, const float* g, const float* b,
                         const float* Xin, float* Ytmp, float* Xout, float* Wp,
                         int CO, int CIN, int CINP,
                         float* mean, float* rstd, hipStream_t stream) {
  pad_weights_kernel<<<(CO * CINP + 255) / 256, 256, 0, stream>>>(W, Wp, CO, CIN, CINP);
  int waves = (CO / 16) * (PP / 64);
  int blocks = (waves * 32 + 255) / 256;
  if (CINP == 132)
    gemm_wmma_kernel<132><<<blocks, 256, 0, stream>>>(Wp, Xin, Ytmp, CO);
  else if (CINP == 68)
    gemm_wmma_kernel<68><<<blocks, 256, 0, stream>>>(Wp, Xin, Ytmp, CO);
  else
    gemm_wmma_kernel<64><<<blocks, 256, 0, stream>>>(Wp, Xin, Ytmp, CO);
  chan_stats_kernel<<<CO, 256, 0, stream>>>(Ytmp, mean, rstd);
  size_t tot = (size_t)CO * PP;
  bn_relu_kernel<<<(unsigned)((tot + 255) / 256), 256, 0, stream>>>(
      Ytmp, mean, rstd, g, b, Xout, CO);
}

extern "C" void kernel_launch(void* const* d_in, const int* in_sizes, int n_in,
                              void* d_out, int out_size, void* d_ws, size_t ws_size,
                              hipStream_t stream) {
  const float* xyz    = (const float*)d_in[0];
  const float* points = (const float*)d_in[1];
  const float* w0  = (const float*)d_in[2];
  const float* g0  = (const float*)d_in[3];
  const float* b0  = (const float*)d_in[4];
  const float* w1  = (const float*)d_in[5];
  const float* g1  = (const float*)d_in[6];
  const float* b1  = (const float*)d_in[7];
  const float* w2  = (const float*)d_in[8];
  const float* g2  = (const float*)d_in[9];
  const float* b2  = (const float*)d_in[10];
  const float* wa0 = (const float*)d_in[11];
  const float* ga0 = (const float*)d_in[12];
  const float* ba0 = (const float*)d_in[13];
  const float* wa1 = (const float*)d_in[14];
  const float* ga1 = (const float*)d_in[15];
  const float* ba1 = (const float*)d_in[16];
  const float* wa2 = (const float*)d_in[17];
  const float* ga2 = (const float*)d_in[18];
  const float* ba2 = (const float*)d_in[19];

  // d_out layout: new_xyz (B*3*S) | new_xyz_1 (B*3*S) | new_points_4 (B*128*S) | gidx (B*S*K ints)
  float* out      = (float*)d_out;
  float* newxyz   = out;
  float* newxyz1  = out + (size_t)BB * 3 * SS;
  float* np4      = out + (size_t)2 * BB * 3 * SS;
  int*   gidx     = (int*)(out + (size_t)2 * BB * 3 * SS + (size_t)BB * 128 * SS);

  // Workspace carve
  char* ws = (char*)d_ws;
  size_t off = 0;
  int* fidx  = (int*)(ws + off);      off = align256(off + (size_t)BB * SS * 4);
  int* gidx1 = (int*)(ws + off);      off = align256(off + (size_t)BB * SS * KK * 4);
  float* mean = (float*)(ws + off);   off = align256(off + 128 * 4);
  float* rstd = (float*)(ws + off);   off = align256(off + 128 * 4);
  float* Wp   = (float*)(ws + off);   off = align256(off + (size_t)128 * 132 * 4);
  float* BIG1 = (float*)(ws + off);   off = align256(off + (size_t)132 * PP * 4);
  float* BIG2 = (float*)(ws + off);   off = align256(off + (size_t)128 * PP * 4);

  // --- Stage 1: FPS + first ball query + grouping ---
  fps_kernel<<<BB, FPS_T, 0, stream>>>(xyz, fidx, newxyz);
  ball_query_kernel<<<(BB * SS * 32 + 255) / 256, 256, 0, stream>>>(xyz, newxyz, gidx);
  group1_kernel<<<(PP + 255) / 256, 256, 0, stream>>>(xyz, points, fidx, gidx, newxyz, BIG1, 132);

  // --- Branch 1 MLP (131 -> 64 -> 64 -> 128) ---
  conv_bn_relu(w0, g0, b0, BIG1, BIG2, BIG1, Wp, 64, 131, 132, mean, rstd, stream);
  conv_bn_relu(w1, g1, b1, BIG1, BIG2, BIG1, Wp, 64, 64, 64, mean, rstd, stream);
  conv_bn_relu(w2, g2, b2, BIG1, BIG2, BIG1, Wp, 128, 64, 64, mean, rstd, stream);

  // --- Attention pooling -> new_xyz_1 ---
  attention_kernel<<<BB * SS, 128, 0, stream>>>(BIG1, gidx, xyz, newxyz1);

  // --- Stage 2: second ball query + grouping (67 channels, pad to 68) ---
  ball_query_kernel<<<(BB * SS * 32 + 255) / 256, 256, 0, stream>>>(xyz, newxyz1, gidx1);
  group2_kernel<<<(PP + 255) / 256, 256, 0, stream>>>(xyz, points, gidx1, newxyz1, BIG2, 68);

  // --- Branch 2 MLP (67 -> 64 -> 64 -> 128) ---
  conv_bn_relu(wa0, ga0, ba0, BIG2, BIG1, BIG2, Wp, 64, 67, 68, mean, rstd, stream);
  conv_bn_relu(wa1, ga1, ba1, BIG2, BIG1, BIG2, Wp, 64, 64, 64, mean, rstd, stream);
  conv_bn_relu(wa2, ga2, ba2, BIG2, BIG1, BIG2, Wp, 128, 64, 64, mean, rstd, stream);

  // --- Final max over K -> new_points_4 ---
  maxk_kernel<<<(BB * 128 * SS + 255) / 256, 256, 0, stream>>>(BIG2, np4);
}